// BehavioralCircuit_566935683778
// MI455X (gfx1250) — compile-verified
//
#include <hip/hip_runtime.h>
#include <hip/hip_bf16.h>

// ---------------------------------------------------------------------------
// BehavioralCircuit: reward-modulated Hebbian recurrence.
//   Rows of W evolve independently -> 512 independent 2-float recurrences.
//   Kernel 1: reward prediction error r[t] (parallel over t).
//   Kernel 2: 16 single-wave workgroups, 1 row per lane, sequential over T.
//             Per-step wave sum via 5x shfl_xor; partial per wave -> scratch.
//   Kernel 3: combine 16 partials per t with V_WMMA_F32_16X16X4_F32
//             (16x16 tile of partials x ones-vector, 4 chained K=4 WMMAs).
// ---------------------------------------------------------------------------

#define LR_CONST 0.1f
#define WINDOW_CONST 10

typedef float v2f __attribute__((ext_vector_type(2)));
typedef float v8f __attribute__((ext_vector_type(8)));

// ---------------- Kernel 1: reward prediction error ------------------------
__global__ void rpe_kernel(const float* __restrict__ rewards,
                           float* __restrict__ r, int T) {
    int t = blockIdx.x * blockDim.x + threadIdx.x;
    if (t >= T) return;
    int start = t - (WINDOW_CONST - 1);
    if (start < 0) start = 0;
    float s = 0.0f;
    for (int k = start; k <= t; ++k) s += rewards[k];
    float cnt = (float)(t - start + 1);
    r[t] = rewards[t] - s / cnt;
}

// ---------------- Kernel 2: the sequential recurrence ----------------------
// grid = NH/32 blocks, block = 32 threads (one wave per WGP).
__global__ __launch_bounds__(32)
void recurrence_kernel(const float* __restrict__ X,
                       const float* __restrict__ r,
                       const float* __restrict__ W0,
                       float* __restrict__ partials, int T) {
    const int lane = threadIdx.x;                 // 0..31
    const int row  = blockIdx.x * 32 + lane;      // hidden unit index
    float w0 = W0[2 * row + 0];
    float w1 = W0[2 * row + 1];
    float* __restrict__ mypart = partials + (size_t)blockIdx.x * (size_t)T;

    const float2* __restrict__ X2 = (const float2*)X;
    const float LOG2E = 1.44269504088896340736f;

    // one-deep software pipeline: loads for t+1 issued before step t's chain
    float2 xv = X2[0];
    float  rt = r[0];
    int t = 0;
    for (; t < T - 1; ++t) {
        float2 xn = X2[t + 1];
        float  rn = r[t + 1];

        float s = fmaf(xv.x, w0, xv.y * w1);
        float e = __builtin_amdgcn_exp2f(-LOG2E * s);
        float h = __builtin_amdgcn_rcpf(1.0f + e);
        float u = (LR_CONST * rt) * h;
        // critical chain: update W immediately
        w0 = fmaf(u, xv.x, w0);
        w1 = fmaf(u, xv.y, w1);
        // off-chain: wave sum of h (32 lanes)
        float p = h;
        p += __shfl_xor(p, 1);
        p += __shfl_xor(p, 2);
        p += __shfl_xor(p, 4);
        p += __shfl_xor(p, 8);
        p += __shfl_xor(p, 16);
        if (lane == 0) mypart[t] = p;

        xv = xn; rt = rn;
    }
    // epilogue: last step
    {
        float s = fmaf(xv.x, w0, xv.y * w1);
        float e = __builtin_amdgcn_exp2f(-LOG2E * s);
        float h = __builtin_amdgcn_rcpf(1.0f + e);
        float p = h;
        p += __shfl_xor(p, 1);
        p += __shfl_xor(p, 2);
        p += __shfl_xor(p, 4);
        p += __shfl_xor(p, 8);
        p += __shfl_xor(p, 16);
        if (lane == 0) mypart[t] = p;
    }
}

// ---------------- Kernel 3: combine partials via WMMA ----------------------
// out[t] = inv * sum_{w=0..15} partials[w*T + t]
// Each wave handles a tile of 16 timesteps: D(16x16) = A(16x16-as-4xK4) x ones.
// A layout per ISA 7.12.2 (32-bit A 16x4): lanes 0-15 carry K=0 (v0) / K=1 (v1),
// lanes 16-31 carry K=2 / K=3. M = timestep within tile, K = partial-row.
__global__ void reduce_kernel(const float* __restrict__ partials,
                              float* __restrict__ out,
                              int T, int nTiles, int nPart, float inv) {
    const int lane        = threadIdx.x & 31;
    const int waveInBlock = threadIdx.x >> 5;
    const int wavesPerBlk = blockDim.x >> 5;
    const int tile        = blockIdx.x * wavesPerBlk + waveInBlock;
    if (tile >= nTiles) return;                 // wave-uniform: EXEC stays all-1
    const int t0 = tile * 16;

    if (nPart == 16 && t0 + 16 <= T) {
        const int m    = lane & 15;             // timestep within tile
        const int kOff = (lane < 16) ? 0 : 2;   // K pair this half-wave owns
        v8f c = {};
        v2f b = {1.0f, 1.0f};                   // ones: layout-irrelevant
        #pragma unroll
        for (int j = 0; j < 4; ++j) {
            v2f a;
            a.x = partials[(size_t)(4 * j + kOff + 0) * (size_t)T + t0 + m];
            a.y = partials[(size_t)(4 * j + kOff + 1) * (size_t)T + t0 + m];
            c = __builtin_amdgcn_wmma_f32_16x16x4_f32(
                    /*neg_a=*/false, a, /*neg_b=*/false, b,
                    /*c_mod=*/(short)0, c, /*reuse_a=*/false, /*reuse_b=*/false);
        }
        // D VGPR v: lane 0 holds D[v][0] (M=v), lane 16 holds D[v+8][0]
        if (lane == 0) {
            #pragma unroll
            for (int v = 0; v < 8; ++v) out[t0 + v] = c[v] * inv;
        } else if (lane == 16) {
            #pragma unroll
            for (int v = 0; v < 8; ++v) out[t0 + 8 + v] = c[v] * inv;
        }
    } else {
        // generic / tail path
        if (lane < 16) {
            int t = t0 + lane;
            if (t < T) {
                float s = 0.0f;
                for (int w = 0; w < nPart; ++w)
                    s += partials[(size_t)w * (size_t)T + t];
                out[t] = s * inv;
            }
        }
    }
}

// ---------------------------------------------------------------------------
extern "C" void kernel_launch(void* const* d_in, const int* in_sizes, int n_in,
                              void* d_out, int out_size, void* d_ws, size_t ws_size,
                              hipStream_t stream) {
    const float* X       = (const float*)d_in[0];   // (T, 2)
    const float* rewards = (const float*)d_in[1];   // (T,)
    const float* W0      = (const float*)d_in[2];   // (NH, 2)

    const int T  = in_sizes[1];
    const int NH = in_sizes[2] / 2;                 // 512
    const int NB = NH / 32;                         // 16 waves / partial rows

    float* r        = (float*)d_ws;                 // T floats
    float* partials = r + T;                        // NB*T floats
    float* out      = (float*)d_out;                // T floats

    // 1) reward prediction error
    rpe_kernel<<<(T + 255) / 256, 256, 0, stream>>>(rewards, r, T);

    // 2) the scan: NB single-wave workgroups, one row per lane
    recurrence_kernel<<<NB, 32, 0, stream>>>(X, r, W0, partials, T);

    // 3) WMMA combine: 16 timesteps per wave, 8 waves per block
    const int nTiles = (T + 15) / 16;
    const int wavesPerBlk = 8;
    const int grid = (nTiles + wavesPerBlk - 1) / wavesPerBlk;
    reduce_kernel<<<grid, wavesPerBlk * 32, 0, stream>>>(
        partials, out, T, nTiles, NB, 1.0f / (float)NH);
}